// HashEncoding_27487790694618
// MI455X (gfx1250) — compile-verified
//
#include <hip/hip_runtime.h>
#include <cstdint>
#include <cstddef>

// ---------------- compile-time constants (match reference) ----------------
#define N_LEVELS 16
#define NPTS     2097152
#define BLOCK    1024
#define PTS_PER_THREAD 4
#define PTS_PER_BLOCK  (BLOCK * PTS_PER_THREAD)   // 4096
#define NBLOCKS        (NPTS / PTS_PER_BLOCK)     // 512

// resolutions = round(16 * 2^(l/3)), l = 0..15
static constexpr int RES[N_LEVELS] = {
    16, 20, 25, 32, 40, 51, 64, 81, 102, 128, 161, 203, 256, 323, 406, 512
};
// tsize = min(4096, res^2); hashed iff tsize < res^2  (levels 7..15)
static constexpr int TSIZE[N_LEVELS] = {
    256, 400, 625, 1024, 1600, 2601, 4096, 4096,
    4096, 4096, 4096, 4096, 4096, 4096, 4096, 4096
};

#define HASH_P1 2654435761u
#define HASH_P2 805459861u

// levels 6..15 staged in LDS: 10 * 4096 float2 = 320 KB (full WGP LDS)
#define N_LDS_LEVELS 10
#define LDS_ENTRIES  (N_LDS_LEVELS * 4096)

struct TablePtrs { const float2* t[N_LEVELS]; };

typedef uint32_t u32x4 __attribute__((ext_vector_type(4)));
typedef uint32_t u32x8 __attribute__((ext_vector_type(8)));

// Issue one TDM DMA: 4096 x 8B elements (32 KB) global -> LDS.
// D# bit layout per CDNA5 ISA ch.8 (group0 128b, group1 256b); 2D tensor so
// descriptor groups 2/3 are NULL -> two-operand tensor_load_to_lds form.
__device__ static inline void tdm_load_table(uint64_t gaddr, uint32_t lds_addr)
{
    u32x4 g0;
    g0.x = 0x1u;                                   // count=1 (valid user D#)
    g0.y = lds_addr;                               // lds_addr [63:32]
    g0.z = (uint32_t)gaddr;                        // global_addr[31:0]
    g0.w = ((uint32_t)(gaddr >> 32) & 0x01FFFFFFu) // global_addr[56:32]
         | 0x80000000u;                            // type=2 ("image") [127:126]
    u32x8 g1;
    g1[0] = 3u << 16;      // wg_mask=0 | data_size=3 (8B) | no flags
    g1[1] = 4096u << 16;   // atomic_barrier_addr=0 | tensor_dim0.lo16 = 4096
    g1[2] = 1u << 16;      // tensor_dim0.hi16=0    | tensor_dim1.lo16 = 1
    g1[3] = 4096u << 16;   // tensor_dim1.hi16=0    | tile_dim0 = 4096
    g1[4] = 1u;            // tile_dim1 = 1 | tile_dim2 = 0
    g1[5] = 4096u;         // tensor_dim0_stride lo32 = 4096
    g1[6] = 4096u << 16;   // stride0 hi16=0 | tensor_dim1_stride lo16 = 4096
    g1[7] = 0u;            // tensor_dim1_stride hi32 = 0
    asm volatile("tensor_load_to_lds %0, %1" :: "s"(g0), "s"(g1) : "memory");
}

__global__ __launch_bounds__(BLOCK)
void HashEncoding_kernel(const float2* __restrict__ coords,
                         TablePtrs tp,
                         float* __restrict__ out)
{
    __shared__ float2 s_tab[LDS_ENTRIES];   // 327680 bytes

    const int tid = threadIdx.x;
    const uint32_t ldsbase = (uint32_t)(uintptr_t)&s_tab[0];

    // ---- level 6 (32 KB): per-wave async global->LDS b128 copies ----
    {
        const uint64_t src = (uint64_t)(uintptr_t)tp.t[6];
        #pragma unroll
        for (int c = 0; c < 2; ++c) {
            const uint32_t goff  = (uint32_t)(tid + c * BLOCK) * 16u;
            const uint32_t laddr = ldsbase + goff;
            asm volatile("global_load_async_to_lds_b128 %0, %1, %2"
                         :: "v"(laddr), "v"(goff), "s"(src)
                         : "memory");
        }
    }

    // ---- levels 7..15 (9 x 32 KB): Tensor Data Mover, issued by wave 0 ----
    if (tid < 32) {
        #pragma unroll
        for (int li = 1; li < N_LDS_LEVELS; ++li) {
            tdm_load_table((uint64_t)(uintptr_t)tp.t[6 + li],
                           ldsbase + (uint32_t)li * 32768u);
        }
        __builtin_amdgcn_s_wait_tensorcnt(0);
    }
    asm volatile("s_wait_asynccnt 0x0" ::: "memory");
    __syncthreads();

    const int pbase = blockIdx.x * PTS_PER_BLOCK + tid;

    #pragma unroll 1
    for (int j = 0; j < PTS_PER_THREAD; ++j) {
        const int p = pbase + j * BLOCK;

        const float2 c = coords[p];
        const float x01 = (c.x + 1.0f) * 0.5f;
        const float y01 = (c.y + 1.0f) * 0.5f;

        float feat[2 * N_LEVELS];

        #pragma unroll
        for (int l = 0; l < N_LEVELS; ++l) {
            const int   res = RES[l];
            const float sx  = x01 * (float)(res - 1);
            const float sy  = y01 * (float)(res - 1);
            const float fsx = floorf(sx);
            const float fsy = floorf(sy);
            const float fx  = sx - fsx;
            const float fy  = sy - fsy;

            // coords in [-1,1) => s in [0, res-1]; floor already in range.
            // Only the +1 neighbor can reach res (top-edge rounding): one min.
            const int x0 = (int)fsx, y0 = (int)fsy;
            const int x1 = min(x0 + 1, res - 1);
            const int y1 = min(y0 + 1, res - 1);

            float2 v00, v01, v10, v11;
            if (l < 6) {
                // small direct-indexed tables: L2/WGP$-resident global gathers
                const float2* __restrict__ t = tp.t[l];
                v00 = t[x0 * res + y0];
                v01 = t[x0 * res + y1];
                v10 = t[x1 * res + y0];
                v11 = t[x1 * res + y1];
            } else if (l == 6) {
                // res 64, tsize 4096 == res^2: direct index, LDS slot 0
                v00 = s_tab[x0 * 64 + y0];
                v01 = s_tab[x0 * 64 + y1];
                v10 = s_tab[x1 * 64 + y0];
                v11 = s_tab[x1 * 64 + y1];
            } else {
                // hashed: (ix*P1 ^ iy*P2) & 4095, table in LDS
                const uint32_t ax0 = (uint32_t)x0 * HASH_P1;
                const uint32_t ax1 = (uint32_t)x1 * HASH_P1;
                const uint32_t by0 = (uint32_t)y0 * HASH_P2;
                const uint32_t by1 = (uint32_t)y1 * HASH_P2;
                const float2* t = s_tab + (l - 6) * 4096;
                v00 = t[(ax0 ^ by0) & 4095u];
                v01 = t[(ax0 ^ by1) & 4095u];
                v10 = t[(ax1 ^ by0) & 4095u];
                v11 = t[(ax1 ^ by1) & 4095u];
            }

            const float gx = 1.0f - fx;
            const float gy = 1.0f - fy;
            const float w00 = gx * gy;
            const float w01 = gx * fy;
            const float w10 = fx * gy;
            const float w11 = fx * fy;

            feat[2 * l + 0] = w00 * v00.x + w01 * v01.x + w10 * v10.x + w11 * v11.x;
            feat[2 * l + 1] = w00 * v00.y + w01 * v01.y + w10 * v10.y + w11 * v11.y;
        }

        // one full 128B row per point -> 8 x b128 stores, whole-line coverage
        float4* __restrict__ orow = (float4*)(out + (size_t)p * (2 * N_LEVELS));
        #pragma unroll
        for (int q = 0; q < 8; ++q) {
            orow[q] = make_float4(feat[4 * q + 0], feat[4 * q + 1],
                                  feat[4 * q + 2], feat[4 * q + 3]);
        }
    }
}

extern "C" void kernel_launch(void* const* d_in, const int* in_sizes, int n_in,
                              void* d_out, int out_size, void* d_ws, size_t ws_size,
                              hipStream_t stream)
{
    (void)in_sizes; (void)out_size; (void)d_ws; (void)ws_size;

    const float2* coords = (const float2*)d_in[0];

    TablePtrs tp;
    if (n_in >= 1 + N_LEVELS) {
        // tables passed as 16 separate inputs (pytree-flattened tuple)
        for (int l = 0; l < N_LEVELS; ++l)
            tp.t[l] = (const float2*)d_in[1 + l];
    } else {
        // fallback: tables concatenated flat in d_in[1]
        const float* base = (const float*)d_in[1];
        size_t off = 0;
        for (int l = 0; l < N_LEVELS; ++l) {
            tp.t[l] = (const float2*)(base + off);
            off += (size_t)TSIZE[l] * 2;
        }
    }

    HashEncoding_kernel<<<NBLOCKS, BLOCK, 0, stream>>>(
        coords, tp, (float*)d_out);
}